// ComplexScaledDotProductAttention_88433376624754
// MI455X (gfx1250) — compile-verified
//
#include <hip/hip_runtime.h>
#include <math.h>

// Complex scaled-dot-product attention for gfx1250 (MI455X).
// fp32 end-to-end using V_WMMA_F32_16X16X4_F32: roofline says the kernel is
// HBM-bound on the 268MB p_attn write (~17us at 23.3TB/s) while all matmuls
// total only ~52 GFLOP, so full-precision f32 WMMA is free. p_attn/out stores
// are non-temporal so the streamed 268MB doesn't evict the L2-resident K/V
// working set (~2MB/head, re-read by 128 workgroups per head).

typedef __attribute__((ext_vector_type(2))) float v2f;
typedef __attribute__((ext_vector_type(8))) float v8f;

#define S_LEN 2048
#define D_HEAD 64
#define BHN 16           // B*H
#define QT 16            // query rows per workgroup
#define LDSTR 2052       // padded LDS row stride in floats (bank-skewed, 8B aligned)
#define NWAVE 8
#define KPW (S_LEN / NWAVE)   // 256 keys per wave

__device__ __forceinline__ v8f wmma4(v2f a, v2f b, v8f c) {
  // D = A(16x4,f32) * B(4x16,f32) + C(16x16,f32)
  return __builtin_amdgcn_wmma_f32_16x16x4_f32(false, a, false, b, (short)0, c,
                                               false, false);
}

extern "C" __global__ __launch_bounds__(256)
void cattn_kernel(const float* __restrict__ Qr, const float* __restrict__ Qi,
                  const float* __restrict__ Kr, const float* __restrict__ Ki,
                  const float* __restrict__ Vr, const float* __restrict__ Vi,
                  float* __restrict__ out) {
  extern __shared__ float smem[];
  float* scr  = smem;                    // [QT][LDSTR]   scores -> exp values
  float* oacc = scr  + QT * LDSTR;       // [2][QT][D_HEAD] cross-wave PV accumulator
  float* part = oacc + 2 * QT * D_HEAD;  // [16][16] reduction partials
  float* rmax = part + 256;              // [16] row max
  float* rinv = rmax + 16;               // [16] 1/row_sum

  const int tid   = threadIdx.x;
  const int lane  = tid & 31;
  const int wv    = tid >> 5;
  const int row16 = lane & 15;  // M (or N) index within 16
  const int half  = lane >> 4;  // selects K-offset +0 / +2 in A/B frags, M+8 in C

  const int bh = blockIdx.x >> 7;            // 128 query tiles per (b,h)
  const int q0 = (blockIdx.x & 127) * QT;

  const float scale = 0.125f;                // 1/sqrt(64)

  const size_t head = (size_t)bh * S_LEN * D_HEAD;
  const float* qrp = Qr + head;
  const float* qip = Qi + head;
  const float* krp = Kr + head;
  const float* kip = Ki + head;
  const float* vrp = Vr + head;
  const float* vip = Vi + head;

  const v8f vzero = {0.f, 0.f, 0.f, 0.f, 0.f, 0.f, 0.f, 0.f};

  // zero cross-wave output accumulator (covered by the phase-1 barrier)
  for (int i = tid; i < 2 * QT * D_HEAD; i += 256) oacc[i] = 0.0f;

  // ---------------- Phase 1: score magnitudes into LDS ----------------
  // Q A-fragments, register resident, reused for all 16 key tiles of this wave.
  v2f qfr[16], qfi[16];
  {
    const float* qb  = qrp + (q0 + row16) * D_HEAD + 2 * half;
    const float* qb2 = qip + (q0 + row16) * D_HEAD + 2 * half;
#pragma unroll
    for (int kk = 0; kk < 16; ++kk) {
      qfr[kk] = *(const v2f*)(qb  + 4 * kk);
      qfi[kk] = *(const v2f*)(qb2 + 4 * kk);
    }
  }

  for (int t = 0; t < 16; ++t) {
    const int kb = wv * KPW + t * 16;
    const float* kbr = krp + (kb + row16) * D_HEAD + 2 * half;
    const float* kbi = kip + (kb + row16) * D_HEAD + 2 * half;
    v8f sre = vzero;  // Qr.Kr + Qi.Ki
    v8f sip = vzero;  // Qr.Ki
    v8f sim = vzero;  // Qi.Kr  (f32 WMMA has no A-negate; subtract after)
#pragma unroll
    for (int kk = 0; kk < 16; ++kk) {
      v2f fr = *(const v2f*)(kbr + 4 * kk);
      v2f fi = *(const v2f*)(kbi + 4 * kk);
      sre = wmma4(qfr[kk], fr, sre);
      sre = wmma4(qfi[kk], fi, sre);
      sip = wmma4(qfr[kk], fi, sip);
      sim = wmma4(qfi[kk], fr, sim);
    }
#pragma unroll
    for (int r = 0; r < 8; ++r) {
      float re  = sre[r];
      float im  = sip[r] - sim[r];
      float mag = scale * __fsqrt_rn(re * re + im * im);
      scr[(r + 8 * half) * LDSTR + kb + row16] = mag;   // C layout: M=r+8*half, N=row16
    }
  }
  __syncthreads();

  // ---------------- Phase 2: softmax over keys ----------------
  {
    const int row = tid >> 4, c0 = tid & 15;
    float m = -3.4e38f;
    for (int j = 0; j < S_LEN / 16; ++j)
      m = fmaxf(m, scr[row * LDSTR + c0 + j * 16]);
    part[row * 16 + c0] = m;
  }
  __syncthreads();
  if (tid < 16) {
    float m = part[tid * 16];
#pragma unroll
    for (int j = 1; j < 16; ++j) m = fmaxf(m, part[tid * 16 + j]);
    rmax[tid] = m;
  }
  __syncthreads();
  {
    const int row = tid >> 4, c0 = tid & 15;
    const float m = rmax[row];
    float s = 0.f;
    for (int j = 0; j < S_LEN / 16; ++j) {
      const int idx = row * LDSTR + c0 + j * 16;
      const float e = __expf(scr[idx] - m);
      scr[idx] = e;                     // keep unnormalized exp in LDS
      s += e;
    }
    part[row * 16 + c0] = s;
  }
  __syncthreads();
  if (tid < 16) {
    float s = 0.f;
#pragma unroll
    for (int j = 0; j < 16; ++j) s += part[tid * 16 + j];
    rinv[tid] = 1.0f / s;
  }
  __syncthreads();

  // stream normalized p_attn to global: dominant HBM traffic, write-once.
  // Non-temporal stores so the 268MB stream doesn't evict K/V from L2.
  {
    float* gP = out + (size_t)2 * BHN * S_LEN * D_HEAD;   // after [2,B,H,S,D]
    for (int idx = tid; idx < QT * S_LEN; idx += 256) {
      const int row = idx >> 11;
      const int col = idx & (S_LEN - 1);
      __builtin_nontemporal_store(
          scr[row * LDSTR + col] * rinv[row],
          &gP[((size_t)(bh * S_LEN + q0 + row)) * S_LEN + col]);
    }
  }

  // ---------------- Phase 3: out = softmax(p) @ V (re & im) ----------------
  v8f aR[4], aI[4];
#pragma unroll
  for (int dt = 0; dt < 4; ++dt) { aR[dt] = vzero; aI[dt] = vzero; }

  const int nbase = wv * KPW;
  for (int ks = 0; ks < KPW / 4; ++ks) {
    const int n0 = nbase + ks * 4;
    // A-fragment of p (exp) tile from LDS: M=row16, K = n0 + v + 2*half
    v2f a = *(const v2f*)&scr[row16 * LDSTR + n0 + 2 * half];
    const float* v0r = vrp + (n0 + 2 * half) * D_HEAD;
    const float* v0i = vip + (n0 + 2 * half) * D_HEAD;
#pragma unroll
    for (int dt = 0; dt < 4; ++dt) {
      const int d = dt * 16 + row16;
      v2f br; br.x = v0r[d]; br.y = v0r[D_HEAD + d];   // K = n0+2h, n0+2h+1
      v2f bi; bi.x = v0i[d]; bi.y = v0i[D_HEAD + d];
      aR[dt] = wmma4(a, br, aR[dt]);
      aI[dt] = wmma4(a, bi, aI[dt]);
    }
  }

  // normalize by 1/rowsum and reduce partials across waves via LDS float atomics
#pragma unroll
  for (int dt = 0; dt < 4; ++dt) {
#pragma unroll
    for (int r = 0; r < 8; ++r) {
      const int m = r + 8 * half;
      const float w = rinv[m];
      atomicAdd(&oacc[m * D_HEAD + dt * 16 + row16], aR[dt][r] * w);
      atomicAdd(&oacc[QT * D_HEAD + m * D_HEAD + dt * 16 + row16], aI[dt][r] * w);
    }
  }
  __syncthreads();

  // final coalesced store of out[2,B,H,S,D] (also streamed, non-temporal)
  for (int idx = tid; idx < 2 * QT * D_HEAD; idx += 256) {
    const int comp = idx >> 10;
    const int rem  = idx & 1023;
    const int m    = rem >> 6;
    const int d    = rem & 63;
    __builtin_nontemporal_store(
        oacc[idx],
        &out[(size_t)comp * (BHN * S_LEN * D_HEAD) +
             ((size_t)(bh * S_LEN + q0 + m)) * D_HEAD + d]);
  }
}

extern "C" void kernel_launch(void* const* d_in, const int* in_sizes, int n_in,
                              void* d_out, int out_size, void* d_ws, size_t ws_size,
                              hipStream_t stream) {
  const float* Qr = (const float*)d_in[0];
  const float* Qi = (const float*)d_in[1];
  const float* Kr = (const float*)d_in[2];
  const float* Ki = (const float*)d_in[3];
  const float* Vr = (const float*)d_in[4];
  const float* Vi = (const float*)d_in[5];
  float* out = (float*)d_out;

  const size_t smem_bytes =
      (size_t)(QT * LDSTR + 2 * QT * D_HEAD + 256 + 16 + 16) * sizeof(float);

  // allow >64KB dynamic LDS (gfx1250 WGP has 320KB)
  (void)hipFuncSetAttribute((const void*)cattn_kernel,
                            hipFuncAttributeMaxDynamicSharedMemorySize,
                            (int)smem_bytes);

  dim3 grid(BHN * (S_LEN / QT));   // 16 * 128 = 2048 workgroups
  dim3 block(256);                 // 8 wave32
  cattn_kernel<<<grid, block, smem_bytes, stream>>>(Qr, Qi, Kr, Ki, Vr, Vi, out);
}